// convLR_154618823203
// MI455X (gfx1250) — compile-verified
//
#include <hip/hip_runtime.h>

// ---------------------------------------------------------------------------
// feature (N=4, C=2048, H=32, W=48); Wc = conv_w[:,:,0,4] -> 2048x2048.
// 94 sequential steps of: state = prelu(Wc @ state_prev + b) + state.
// Per step: GEMM M=2048, N=128, K=2048 (fp32, V_WMMA_F32_16X16X4_F32),
// split-K=4 across blocks; A fragments staged global->LDS (async, double
// buffered) and shared by all 8 waves of the block.
// ---------------------------------------------------------------------------
#define C_DIM   2048
#define N_DIM   4
#define H_DIM   32
#define W_DIM   48
#define NH      128                 // GEMM N dimension (N*H)
#define KSTEPS  (C_DIM / 4)         // 512 k-steps of depth 4
#define KCH     4                   // split-K factor
#define KCH_STEPS (KSTEPS / KCH)    // 128 k-steps per K-chunk
#define KC      16                  // k-steps per LDS staging chunk
#define NCHUNK  (KCH_STEPS / KC)    // 8 staged chunks per kernel
#define CHUNK_FLTS (8 * KC * 64)    // floats per staged chunk (8192 = 32KB)
#define MTILES  (C_DIM / 16)        // 128
#define SLICE   (C_DIM * NH)        // floats per state slice (262144)

typedef __attribute__((ext_vector_type(2))) float v2f;
typedef __attribute__((ext_vector_type(8))) float v8f;
typedef __attribute__((ext_vector_type(4))) int   v4i;

#if defined(__has_builtin)
#if __has_builtin(__builtin_amdgcn_global_load_async_to_lds_b128)
#define HAS_ASYNC_LDS 1
#endif
#endif
#ifndef HAS_ASYNC_LDS
#define HAS_ASYNC_LDS 0
#endif

// State layout: (even,odd)-row float pairs so B fragment (B[k][n],B[k+1][n])
// with k even is one aligned 8-byte load. Flat float offset of (row c, col n):
__device__ __forceinline__ int stIdx(int c, int n) {
    return (((c >> 1) * NH) + n) * 2 + (c & 1);
}

// ---------------------------------------------------------------------------
// Pre-shuffle Wc into 16x4 f32 A-fragment lane layout:
//   lane L holds A[M=L%16][K=2*(L/16)], A[M][K+1]
// Apre[((mTile*KSTEPS)+kstep)*32 + lane] : v2f. For one mTile, consecutive
// ksteps are contiguous -> 4KB contiguous segments for LDS staging.
// ---------------------------------------------------------------------------
__global__ void prep_w_kernel(const float* __restrict__ conv_w,
                              float* __restrict__ Apre) {
    int gid   = blockIdx.x * blockDim.x + threadIdx.x;
    int lane  = gid & 31;
    int kstep = (gid >> 5) & (KSTEPS - 1);
    int mTile = gid >> 14;
    int m = mTile * 16 + (lane & 15);
    int k = kstep * 4 + ((lane >> 4) << 1);
    v2f val;
    val.x = conv_w[(size_t)(m * C_DIM + k    ) * 9 + 4];
    val.y = conv_w[(size_t)(m * C_DIM + k + 1) * 9 + 4];
    *reinterpret_cast<v2f*>(Apre + (size_t)gid * 2) = val;
}

// feature (N,C,H,W) -> state FL[w] (C x NH paired layout), col = n*32+h
__global__ void prep_x_kernel(const float* __restrict__ feature,
                              float* __restrict__ FL) {
    size_t idx = (size_t)blockIdx.x * blockDim.x + threadIdx.x;
    const size_t total = (size_t)W_DIM * SLICE;
    if (idx >= total) return;
    int j = (int)(idx % NH);
    int c = (int)((idx / NH) % C_DIM);
    int w = (int)(idx / ((size_t)NH * C_DIM));
    int n = j >> 5, h = j & 31;
    float v = feature[(((size_t)n * C_DIM + c) * H_DIM + h) * W_DIM + w];
    FL[(size_t)w * SLICE + stIdx(c, j)] = v;
}

// ---------------------------------------------------------------------------
// Split-K GEMM partial. Block (bx, kc): rows [bx*128, bx*128+128) over K
// chunk kc; writes its own partial slice Tpart[kc] (row-major 2048x128).
// 64 blocks x 8 waves; wave tile 64(M) x 32(N) = 4x2 WMMA tiles.
// A fragments: global -> LDS (async b128, double buffered), ds_load_b64 reads.
// ---------------------------------------------------------------------------
__global__ __launch_bounds__(256) void gemm_partial_kernel(
    const float* __restrict__ Apre, const float* __restrict__ src,
    float* __restrict__ Tpart) {

    __shared__ float smem[2 * CHUNK_FLTS];          // 64 KB, double buffered

    const int lane  = threadIdx.x & 31;
    const int wid   = threadIdx.x >> 5;             // 0..7
    const int wm    = wid & 1;
    const int wn    = wid >> 1;                     // 0..3
    const int Mbase = blockIdx.x * 128 + wm * 64;
    const int Nbase = wn * 32;
    const int kc    = blockIdx.y;                   // 0..3
    const int mT0   = (blockIdx.x * 128) >> 4;      // first mTile of block
    const unsigned kcBase = (unsigned)kc * KCH_STEPS;

    v8f acc[4][2] = {};

    // B element offsets into paired state slice (advance 512 floats / k-step).
    const int k0 = kc * (KCH_STEPS * 4) + ((lane >> 4) << 1);   // even
    unsigned bOff0 = (unsigned)stIdx(k0, Nbase + (lane & 15));
    unsigned bOff1 = (unsigned)stIdx(k0, Nbase + 16 + (lane & 15));

    // ---- stage chunk c of this block's A fragments into LDS buffer buf ----
    auto stage = [&](int c, int buf) {
#if HAS_ASYNC_LDS
        // wave 'wid' stages the 4KB contiguous segment of mTile (mT0+wid)
        unsigned gseg = ((unsigned)(mT0 + wid) * KSTEPS + kcBase
                         + (unsigned)c * KC) * 64u;
        unsigned lseg = (unsigned)buf * CHUNK_FLTS + (unsigned)wid * (KC * 64);
#pragma unroll
        for (int i = 0; i < 8; ++i) {               // 8 x 512B per wave
            const float* gp = Apre + gseg + (unsigned)i * 128u
                            + (unsigned)lane * 4u;
            const float* lp = smem + lseg + (unsigned)i * 128u
                            + (unsigned)lane * 4u;
            __builtin_amdgcn_global_load_async_to_lds_b128(
                (v4i*)gp,
                (__attribute__((address_space(3))) v4i*)
                    (unsigned)(unsigned long long)lp,
                0, 0);
        }
#else
        // fallback: regular load + ds_store staging (16 v2f per thread)
        const unsigned t = (unsigned)threadIdx.x;
#pragma unroll
        for (int j = 0; j < 16; ++j) {
            unsigned v    = (unsigned)j * 256u + t;     // v2f index in chunk
            unsigned mL   = v >> 9;                     // 512 v2f per segment
            unsigned rem  = v & 511u;
            unsigned gOff = ((mT0 + mL) * KSTEPS + kcBase + (unsigned)c * KC)
                            * 64u + rem * 2u;
            unsigned lOff = (unsigned)buf * CHUNK_FLTS + mL * (KC * 64)
                            + rem * 2u;
            *reinterpret_cast<v2f*>(smem + lOff) =
                *reinterpret_cast<const v2f*>(Apre + gOff);
        }
#endif
    };

    stage(0, 0);

    for (int c = 0; c < NCHUNK; ++c) {
#if HAS_ASYNC_LDS
        asm volatile("s_wait_asynccnt 0" ::: "memory");
#endif
        __syncthreads();                            // chunk c visible to all
        if (c + 1 < NCHUNK) stage(c + 1, (c + 1) & 1);

        const unsigned lbase = (unsigned)(c & 1) * CHUNK_FLTS
                             + (unsigned)(wm * 4) * (KC * 64)
                             + (unsigned)lane * 2u;
#pragma unroll 2
        for (int ks = 0; ks < KC; ++ks) {
            v2f a[4], b[2];
#pragma unroll
            for (int t = 0; t < 4; ++t)             // ds_load_b64, bank-free
                a[t] = *reinterpret_cast<const v2f*>(
                    smem + lbase + (unsigned)t * (KC * 64) + (unsigned)ks * 64u);
            b[0] = *reinterpret_cast<const v2f*>(src + bOff0);
            b[1] = *reinterpret_cast<const v2f*>(src + bOff1);
            bOff0 += 512u;
            bOff1 += 512u;
#pragma unroll
            for (int mt = 0; mt < 4; ++mt)
#pragma unroll
                for (int nt = 0; nt < 2; ++nt)
                    acc[mt][nt] = __builtin_amdgcn_wmma_f32_16x16x4_f32(
                        false, a[mt], false, b[nt], (short)0, acc[mt][nt],
                        false, false);
        }
        __syncthreads();                            // done reading buf (c&1)
    }

    // Write this K-chunk's partial slice (fully overwritten -> deterministic).
    float* outp = Tpart + (size_t)kc * SLICE;
#pragma unroll
    for (int mt = 0; mt < 4; ++mt) {
#pragma unroll
        for (int nt = 0; nt < 2; ++nt) {
            const int n = Nbase + nt * 16 + (lane & 15);
#pragma unroll
            for (int r = 0; r < 8; ++r) {
                const int row = Mbase + mt * 16 + ((lane >> 4) << 3) + r;
                outp[(unsigned)(row * NH + n)] = acc[mt][nt][r];
            }
        }
    }
}

// Sum 4 K-chunk partials + bias, PReLU, residual-add in place on state slice.
__global__ void step_epilogue_kernel(const float* __restrict__ Tpart,
                                     float* __restrict__ dst,
                                     const float* __restrict__ bias,
                                     const float* __restrict__ alpha_p) {
    const unsigned idx = blockIdx.x * blockDim.x + threadIdx.x; // 0..SLICE-1
    const int n   = (int)(idx & (NH - 1));
    const int row = (int)(idx >> 7);
    float y = Tpart[idx] + Tpart[idx + SLICE] + Tpart[idx + 2u * SLICE]
            + Tpart[idx + 3u * SLICE] + bias[row];
    const float alpha = *alpha_p;
    y = (y >= 0.0f) ? y : alpha * y;
    const int o = stIdx(row, n);
    dst[o] = y + dst[o];
}

// state FL -> out (N,C,H,W)
__global__ void finalize_kernel(const float* __restrict__ FL,
                                float* __restrict__ out) {
    size_t idx = (size_t)blockIdx.x * blockDim.x + threadIdx.x;
    const size_t total = (size_t)N_DIM * C_DIM * H_DIM * W_DIM;
    if (idx >= total) return;
    int w = (int)(idx % W_DIM);
    int h = (int)((idx / W_DIM) % H_DIM);
    int c = (int)((idx / (W_DIM * H_DIM)) % C_DIM);
    int n = (int)(idx / ((size_t)W_DIM * H_DIM * C_DIM));
    out[idx] = FL[(size_t)w * SLICE + stIdx(c, n * H_DIM + h)];
}

// ---------------------------------------------------------------------------
extern "C" void kernel_launch(void* const* d_in, const int* in_sizes, int n_in,
                              void* d_out, int out_size, void* d_ws,
                              size_t ws_size, hipStream_t stream) {
    (void)in_sizes; (void)n_in; (void)out_size; (void)ws_size;
    const float* feature = (const float*)d_in[0];
    const float* conv_w  = (const float*)d_in[1];
    const float* conv_b  = (const float*)d_in[2];
    const float* prelu_a = (const float*)d_in[3];
    float* out = (float*)d_out;

    // ws: [0,16MB) A fragments | [16,64MB) state FL | [64,68MB) K partials
    float* Apre  = (float*)d_ws;
    float* FL    = Apre + (size_t)C_DIM * C_DIM;
    float* Tpart = FL + (size_t)W_DIM * SLICE;

    prep_w_kernel<<<(MTILES * KSTEPS * 32) / 256, 256, 0, stream>>>(conv_w, Apre);

    const size_t xTot = (size_t)W_DIM * SLICE;
    prep_x_kernel<<<(unsigned)((xTot + 255) / 256), 256, 0, stream>>>(feature, FL);

    const dim3 pg(C_DIM / 128, KCH);
    const unsigned eg = SLICE / 256;

    // Forward: fl[w] = prelu(Wc@fl[w-1]+b) + x[w]
    for (int w = 1; w < W_DIM; ++w) {
        gemm_partial_kernel<<<pg, 256, 0, stream>>>(
            Apre, FL + (size_t)(w - 1) * SLICE, Tpart);
        step_epilogue_kernel<<<eg, 256, 0, stream>>>(
            Tpart, FL + (size_t)w * SLICE, conv_b, prelu_a);
    }
    // Backward: out[w] = prelu(Wc@out[w+1]+b) + fl[w]
    for (int w = W_DIM - 2; w >= 0; --w) {
        gemm_partial_kernel<<<pg, 256, 0, stream>>>(
            Apre, FL + (size_t)(w + 1) * SLICE, Tpart);
        step_epilogue_kernel<<<eg, 256, 0, stream>>>(
            Tpart, FL + (size_t)w * SLICE, conv_b, prelu_a);
    }

    const size_t oTot = (size_t)N_DIM * C_DIM * H_DIM * W_DIM;
    finalize_kernel<<<(unsigned)((oTot + 255) / 256), 256, 0, stream>>>(FL, out);
}